// HaarConv2D_9732395893263
// MI455X (gfx1250) — compile-verified
//
#include <hip/hip_runtime.h>

// Haar 2x2 decimating transform, B=16, C=64, H=512, W=512, fp32.
// Roofline: 1.61 GB mandatory traffic, AI ~0.25 FLOP/B -> pure HBM stream,
// ~69 us floor @ 23.3 TB/s. WMMA is useless here; the CDNA5 leverage is the
// data-movement path:
//   - GLOBAL_LOAD_ASYNC_TO_LDS_B128 (ASYNCcnt): VGPR-free loads, per-WGP
//     request queue -> more outstanding HBM requests than per-wave loads.
//   - ds_load_b128 read-back from a padded (80B stride) layout: hits the
//     2-clk/wave LDS minimum, no bank-conflict serialization.
//   - global_store_b128 th:TH_STORE_NT: streamed output, no L2 pollution.

typedef __attribute__((ext_vector_type(4))) float f32x4;

#define PLANES      (16 * 64)          // 1024
#define IN_W        512
#define OUT_H       256
#define OUT_W       256
#define PLANE_IN    (512 * 512)
#define PLANE_OUT   (256 * 256)
#define THR_PER_ROW (OUT_W / 4)        // 64 threads/row, 4 outputs each
#define BLOCK       256
#define CHUNK       80                 // 64B staged data + 16B pad (16B aligned)

__global__ __launch_bounds__(BLOCK) void haar2x2_async_kernel(
    const float* __restrict__ x,
    float* __restrict__ low,
    float* __restrict__ det)
{
    __shared__ __align__(16) unsigned char smem[BLOCK * CHUNK];   // 20 KB

    const int tid = threadIdx.x;
    const int t   = blockIdx.x * BLOCK + tid;

    const int lane_in_row = t & (THR_PER_ROW - 1);   // 0..63
    const int orow        = (t >> 6) & (OUT_H - 1);  // 0..255
    const int plane       = t >> 14;                 // 0..1023

    const int in_base  = plane * PLANE_IN + (2 * orow) * IN_W + lane_in_row * 8;
    const int out_base = plane * PLANE_OUT + orow * OUT_W + lane_in_row * 4;

    const float* g0 = x + in_base;          // even input row, 32B
    const float* g1 = x + in_base + IN_W;   // odd  input row, 32B

    // 32-bit LDS byte offset for the async-load VDST operand: generic
    // pointers to LDS are {aperture_hi32, lds_offset32}, so truncation
    // yields the DS address.
    unsigned char* lp = &smem[tid * CHUNK];
    const unsigned lds0 = (unsigned)(unsigned long long)lp;        // rows 0 data
    const unsigned lds1 = lds0 + 32;                               // rows 1 data

    // Stage 64B/lane. INST_OFFSET applies to both the global and LDS side,
    // so offset:16 fills the second 16B half with the same VGPR operands.
    asm volatile(
        "global_load_async_to_lds_b128 %0, %2, off th:TH_LOAD_NT\n\t"
        "global_load_async_to_lds_b128 %0, %2, off offset:16 th:TH_LOAD_NT\n\t"
        "global_load_async_to_lds_b128 %1, %3, off th:TH_LOAD_NT\n\t"
        "global_load_async_to_lds_b128 %1, %3, off offset:16 th:TH_LOAD_NT"
        :
        : "v"(lds0), "v"(lds1), "v"(g0), "v"(g1)
        : "memory");

    // Each lane reads back only bytes it staged itself -> own ASYNCcnt is a
    // sufficient ordering point, no workgroup barrier required.
    asm volatile("s_wait_asynccnt 0x0" ::: "memory");

    const f32x4* lv = (const f32x4*)lp;    // 4x ds_load_b128
    const f32x4 r0a = lv[0];
    const f32x4 r0b = lv[1];
    const f32x4 r1a = lv[2];
    const f32x4 r1b = lv[3];

    // Butterfly: a=even/even, b=even/odd, c=odd/even, d=odd/odd
    // low = 0.5*((a+d)+(b+c)),  det = 0.5*((a+d)-(b+c))
    f32x4 lo, de;
    {
        const float p = r0a.x + r1a.y, q = r0a.y + r1a.x;
        lo.x = 0.5f * (p + q);  de.x = 0.5f * (p - q);
    }
    {
        const float p = r0a.z + r1a.w, q = r0a.w + r1a.z;
        lo.y = 0.5f * (p + q);  de.y = 0.5f * (p - q);
    }
    {
        const float p = r0b.x + r1b.y, q = r0b.y + r1b.x;
        lo.z = 0.5f * (p + q);  de.z = 0.5f * (p - q);
    }
    {
        const float p = r0b.z + r1b.w, q = r0b.w + r1b.z;
        lo.w = 0.5f * (p + q);  de.w = 0.5f * (p - q);
    }

    __builtin_nontemporal_store(lo, (f32x4*)(low + out_base));
    __builtin_nontemporal_store(de, (f32x4*)(det + out_base));
}

extern "C" void kernel_launch(void* const* d_in, const int* in_sizes, int n_in,
                              void* d_out, int out_size, void* d_ws, size_t ws_size,
                              hipStream_t stream)
{
    (void)in_sizes; (void)n_in; (void)out_size; (void)d_ws; (void)ws_size;

    const float* x = (const float*)d_in[0];
    float* low = (float*)d_out;                        // tuple element 0
    float* det = low + (size_t)PLANES * PLANE_OUT;     // tuple element 1

    const int total_threads = PLANES * OUT_H * THR_PER_ROW;   // 16,777,216
    const int grid = total_threads / BLOCK;                   // 65,536

    hipLaunchKernelGGL(haar2x2_async_kernel, dim3(grid), dim3(BLOCK), 0, stream,
                       x, low, det);
}